// get_model_47184510714284
// MI455X (gfx1250) — compile-verified
//
#include <hip/hip_runtime.h>
#include <math.h>

typedef __attribute__((ext_vector_type(2))) float v2f;
typedef __attribute__((ext_vector_type(8))) float v8f;

#define BATCH 8
#define NPTS  4096
#define NCLS  13
#define EPSBN 1e-5f

// ---------------------------------------------------------------------------
// Generic batched WMMA f32 GEMM:  C[M x N] = A[M x K] * B[K x N]
// All row-major. BT=true means B is stored [N x K] row-major (B transposed).
// One wave per block computes a 16(M) x 64(N) strip: the A fragment is loaded
// once per K-step and reused by 4 v_wmma_f32_16x16x4_f32 issues.
// Interior tiles take an unguarded fast path (no exec-mask churn); only the
// last tile row/col (or ragged K) takes the predicated edge path.
// Fragment mapping per CDNA5 ISA 7.12.2:
//   A 16x4 : lane l<16 holds A[l][k0],A[l][k0+1]; lane l>=16 holds A[l-16][k0+2..3]
//   B 4x16 : lane l<16 holds B[k0][l],B[k0+1][l]; lane l>=16 holds B[k0+2..3][l-16]
//   C 16x16: vgpr v, lane l<16 -> C[v][l]; lane l>=16 -> C[v+8][l-16]
// ---------------------------------------------------------------------------
template <bool BT>
__global__ void wmma_gemm(const float* __restrict__ A, const float* __restrict__ Bm,
                          float* __restrict__ C, int M, int N, int K,
                          int lda, int ldb, int ldc,
                          long sA, long sB, long sC)
{
  const float* Ab = A  + (long)blockIdx.z * sA;
  const float* Bb = Bm + (long)blockIdx.z * sB;
  float*       Cb = C  + (long)blockIdx.z * sC;
  int lane  = threadIdx.x & 31;
  int row16 = lane & 15;
  int kq    = lane >> 4;          // 0 or 1
  int tM    = blockIdx.y * 16;
  int tN    = blockIdx.x * 64;
  v8f acc0 = {}, acc1 = {}, acc2 = {}, acc3 = {};
  int aRow = tM + row16;
  bool interior = (tM + 16 <= M) && (tN + 64 <= N) && ((K & 3) == 0);

  if (interior) {
    const float* Ap = Ab + (long)aRow * lda + 2 * kq;
    int col0 = tN + row16;
    for (int k0 = 0; k0 < K; k0 += 4) {
      v2f a;
      a.x = Ap[k0];
      a.y = Ap[k0 + 1];
      int ka = k0 + 2 * kq;
#pragma unroll
      for (int t = 0; t < 4; ++t) {
        int col = col0 + t * 16;
        v2f bv;
        if (BT) {
          const float* Bp = Bb + (long)col * ldb + ka;
          bv.x = Bp[0];
          bv.y = Bp[1];
        } else {
          bv.x = Bb[(long)(ka    ) * ldb + col];
          bv.y = Bb[(long)(ka + 1) * ldb + col];
        }
        v8f& ac = (t == 0) ? acc0 : (t == 1) ? acc1 : (t == 2) ? acc2 : acc3;
        ac = __builtin_amdgcn_wmma_f32_16x16x4_f32(false, a, false, bv,
                                                   (short)0, ac, false, false);
      }
    }
#pragma unroll
    for (int t = 0; t < 4; ++t) {
      v8f ac = (t == 0) ? acc0 : (t == 1) ? acc1 : (t == 2) ? acc2 : acc3;
      int col = col0 + t * 16;
#pragma unroll
      for (int v = 0; v < 8; ++v) {
        int r = tM + v + 8 * kq;
        Cb[(long)r * ldc + col] = ac[v];
      }
    }
  } else {
    bool aOk = aRow < M;
    for (int k0 = 0; k0 < K; k0 += 4) {
      int ka = k0 + 2 * kq;
      v2f a;
      a.x = (aOk && ka     < K) ? Ab[(long)aRow * lda + ka    ] : 0.f;
      a.y = (aOk && ka + 1 < K) ? Ab[(long)aRow * lda + ka + 1] : 0.f;
#pragma unroll
      for (int t = 0; t < 4; ++t) {
        int col = tN + t * 16 + row16;
        v2f bv;
        if (BT) {
          bv.x = (col < N && ka     < K) ? Bb[(long)col * ldb + ka    ] : 0.f;
          bv.y = (col < N && ka + 1 < K) ? Bb[(long)col * ldb + ka + 1] : 0.f;
        } else {
          bv.x = (col < N && ka     < K) ? Bb[(long)(ka    ) * ldb + col] : 0.f;
          bv.y = (col < N && ka + 1 < K) ? Bb[(long)(ka + 1) * ldb + col] : 0.f;
        }
        v8f& ac = (t == 0) ? acc0 : (t == 1) ? acc1 : (t == 2) ? acc2 : acc3;
        ac = __builtin_amdgcn_wmma_f32_16x16x4_f32(false, a, false, bv,
                                                   (short)0, ac, false, false);
      }
    }
#pragma unroll
    for (int t = 0; t < 4; ++t) {
      v8f ac = (t == 0) ? acc0 : (t == 1) ? acc1 : (t == 2) ? acc2 : acc3;
      int col = tN + t * 16 + row16;
      if (col < N) {
#pragma unroll
        for (int v = 0; v < 8; ++v) {
          int r = tM + v + 8 * kq;
          if (r < M) Cb[(long)r * ldc + col] = ac[v];
        }
      }
    }
  }
}

// ---------------------------------------------------------------------------
// FPS: one block per batch; dist[] lives in LDS; block argmax w/ lowest-index
// tie break (matches jnp.argmax).
// ---------------------------------------------------------------------------
__global__ void fps_kernel(const float* __restrict__ coords, int n, int npoint,
                           int* __restrict__ outIdx)
{
  __shared__ float dist[4096];
  __shared__ float rv[256];
  __shared__ int   ri[256];
  __shared__ int   sFar;
  int b = blockIdx.x;
  const float* xb = coords + (long)b * n * 3;
  int t = threadIdx.x;
  for (int i = t; i < n; i += 256) dist[i] = 1e10f;
  if (t == 0) sFar = 0;
  __syncthreads();
  for (int it = 0; it < npoint; ++it) {
    int far = sFar;
    if (t == 0) outIdx[b * npoint + it] = far;
    float cx = xb[far*3], cy = xb[far*3+1], cz = xb[far*3+2];
    float bm = -1.f; int bi = 0x7fffffff;
    for (int i = t; i < n; i += 256) {
      float dx = xb[i*3] - cx, dy = xb[i*3+1] - cy, dz = xb[i*3+2] - cz;
      float d = dx*dx + dy*dy + dz*dz;
      float dd = fminf(dist[i], d);
      dist[i] = dd;
      if (dd > bm) { bm = dd; bi = i; }
    }
    rv[t] = bm; ri[t] = bi;
    __syncthreads();
    for (int sft = 128; sft > 0; sft >>= 1) {
      if (t < sft) {
        if (rv[t+sft] > rv[t] || (rv[t+sft] == rv[t] && ri[t+sft] < ri[t])) {
          rv[t] = rv[t+sft]; ri[t] = ri[t+sft];
        }
      }
      __syncthreads();
    }
    if (t == 0) sFar = ri[0];
    __syncthreads();
  }
}

// ---------------------------------------------------------------------------
// small elementwise / reduction kernels
// ---------------------------------------------------------------------------
__global__ void k_strip_xyz(const float* __restrict__ in, float* __restrict__ out, long total)
{
  long tid = (long)blockIdx.x * 256 + threadIdx.x; if (tid >= total) return;
  int c = tid % 3; long r = tid / 3; int n = r % NPTS; int b = r / NPTS;
  out[tid] = in[((long)b * 9 + c) * NPTS + n];
}

__global__ void k_gather_coords(const float* __restrict__ coords, const int* __restrict__ idx,
                                int n, int np, float* __restrict__ out, long total)
{
  long tid = (long)blockIdx.x * 256 + threadIdx.x; if (tid >= total) return;
  int c = tid % 3; long r = tid / 3; int s = r % np; int b = r / np;
  out[tid] = coords[((long)b * n + idx[(long)b * np + s]) * 3 + c];
}

__global__ void k_ball(const float* __restrict__ coords, const float* __restrict__ newc,
                       int n, int np, int kk, float r2, int* __restrict__ out, long total)
{
  long tid = (long)blockIdx.x * 256 + threadIdx.x; if (tid >= total) return;
  int b = tid / np, s = tid % np;
  const float* nc = &newc[((long)b * np + s) * 3];
  int* o = &out[((long)b * np + s) * kk];
  int cnt = 0, first = 0;
  for (int j = 0; j < n && cnt < kk; ++j) {
    float dx = coords[((long)b*n + j)*3    ] - nc[0];
    float dy = coords[((long)b*n + j)*3 + 1] - nc[1];
    float dz = coords[((long)b*n + j)*3 + 2] - nc[2];
    float d = dx*dx + dy*dy + dz*dz;
    if (d <= r2) { if (cnt == 0) first = j; o[cnt++] = j; }
  }
  for (; cnt < kk; ++cnt) o[cnt] = first;
}

__global__ void k_group(const float* __restrict__ coords, const float* __restrict__ newc,
                        const float* __restrict__ feats, const int* __restrict__ ballIdx,
                        int n, int np, int kk, int Cf, float* __restrict__ out, long total)
{
  long tid = (long)blockIdx.x * 256 + threadIdx.x; if (tid >= total) return;
  int C = 3 + Cf;
  int s = tid % np; long r = tid / np; int q = r % kk; r /= kk; int c = r % C; int b = r / C;
  int j = ballIdx[((long)b * np + s) * kk + q];
  float v;
  if (c < 3) v = coords[((long)b*n + j)*3 + c] - newc[((long)b*np + s)*3 + c];
  else       v = feats[((long)b*Cf + (c-3))*n + j];
  out[tid] = v;
}

__global__ void bn_stats(const float* __restrict__ z, int C, long Msp, long bStride,
                         float* __restrict__ mv)
{
  int c = blockIdx.x;
  __shared__ float s1[256], s2[256];
  float a = 0.f, b2 = 0.f;
  for (long i = threadIdx.x; i < (long)BATCH * Msp; i += 256) {
    int bb = i / Msp; long sp = i % Msp;
    float v = z[(long)bb * bStride + (long)c * Msp + sp];
    a += v; b2 += v * v;
  }
  s1[threadIdx.x] = a; s2[threadIdx.x] = b2;
  __syncthreads();
  for (int sft = 128; sft > 0; sft >>= 1) {
    if (threadIdx.x < sft) { s1[threadIdx.x] += s1[threadIdx.x+sft]; s2[threadIdx.x] += s2[threadIdx.x+sft]; }
    __syncthreads();
  }
  if (threadIdx.x == 0) {
    float inv = 1.f / (float)((long)BATCH * Msp);
    float m = s1[0] * inv;
    mv[2*c] = m; mv[2*c+1] = s2[0] * inv - m * m;
  }
}

__global__ void k_bn_apply_relu(const float* __restrict__ z, const float* __restrict__ mv,
                                const float* __restrict__ g, const float* __restrict__ be,
                                int C, long Msp, float* __restrict__ out, long total)
{
  long tid = (long)blockIdx.x * 256 + threadIdx.x; if (tid >= total) return;
  int c = (int)((tid / Msp) % C);
  float m = mv[2*c], v = mv[2*c+1];
  float y = g[c] * (z[tid] - m) * rsqrtf(v + EPSBN) + be[c];
  out[tid] = fmaxf(y, 0.f);
}

__global__ void k_maxpool(const float* __restrict__ z, int C, int kk, int np,
                          float* __restrict__ out, long total)
{
  long tid = (long)blockIdx.x * 256 + threadIdx.x; if (tid >= total) return;
  int s = tid % np; long r = tid / np; int c = r % C; int b = r / C;
  const float* base = z + (((long)b * C + c) * kk) * np + s;
  float m = -1e30f;
  for (int q = 0; q < kk; ++q) m = fmaxf(m, base[(long)q * np]);
  out[tid] = m;
}

__global__ void k_cm2sd(const float* __restrict__ in, int C, int n,
                        float* __restrict__ out, long total)
{
  long tid = (long)blockIdx.x * 256 + threadIdx.x; if (tid >= total) return;
  int i = tid % n; long r = tid / n; int c = r % C; int b = r / C;
  out[((long)b * n + i) * C + c] = in[tid];
}

__global__ void k_sd2cm(const float* __restrict__ in, int C, int n,
                        float* __restrict__ out, long total)
{
  long tid = (long)blockIdx.x * 256 + threadIdx.x; if (tid >= total) return;
  int i = tid % n; long r = tid / n; int c = r % C; int b = r / C;
  out[tid] = in[((long)b * n + i) * C + c];
}

__global__ void k_ln(const float* __restrict__ in, const float* __restrict__ g,
                     const float* __restrict__ bb, int rows, int d, float* __restrict__ out)
{
  int r = blockIdx.x * 256 + threadIdx.x; if (r >= rows) return;
  const float* x = in + (long)r * d;
  float m = 0.f;
  for (int c = 0; c < d; ++c) m += x[c];
  m /= d;
  float v = 0.f;
  for (int c = 0; c < d; ++c) { float t = x[c] - m; v += t * t; }
  v /= d;
  float inv = rsqrtf(v + EPSBN);
  float* o = out + (long)r * d;
  for (int c = 0; c < d; ++c) o[c] = g[c] * (x[c] - m) * inv + bb[c];
}

__global__ void k_split_heads(const float* __restrict__ qkv, int s,
                              float* __restrict__ q, float* __restrict__ k,
                              float* __restrict__ v, long total)
{
  long tid = (long)blockIdx.x * 256 + threadIdx.x; if (tid >= total) return;
  int dh = tid % 64; long r = tid / 64; int i = r % s; r /= s; int h = r % 8; int b = r / 8;
  long base = ((long)b * s + i) * 1536 + h * 64 + dh;
  q[tid] = qkv[base];
  k[tid] = qkv[base + 512];
  v[tid] = qkv[base + 1024];
}

__global__ void k_softmax(float* __restrict__ att, int rows, int len, float scale)
{
  int r = blockIdx.x * 256 + threadIdx.x; if (r >= rows) return;
  float* x = att + (long)r * len;
  float m = -1e30f;
  for (int j = 0; j < len; ++j) m = fmaxf(m, x[j] * scale);
  float s = 0.f;
  for (int j = 0; j < len; ++j) s += expf(x[j] * scale - m);
  float inv = 1.f / s;
  for (int j = 0; j < len; ++j) x[j] = expf(x[j] * scale - m) * inv;
}

__global__ void k_merge_heads(const float* __restrict__ o, int s,
                              float* __restrict__ osd, long total)
{
  long tid = (long)blockIdx.x * 256 + threadIdx.x; if (tid >= total) return;
  int dh = tid % 64; long r = tid / 64; int i = r % s; r /= s; int h = r % 8; int b = r / 8;
  osd[((long)b * s + i) * 512 + h * 64 + dh] = o[tid];
}

__global__ void k_add_bias_res(float* __restrict__ x, const float* __restrict__ p,
                               const float* __restrict__ bias, int d, long total)
{
  long tid = (long)blockIdx.x * 256 + threadIdx.x; if (tid >= total) return;
  x[tid] = x[tid] + p[tid] + bias[tid % d];
}

__global__ void k_gelu_bias(float* __restrict__ x, const float* __restrict__ bias,
                            int c, long total)
{
  long tid = (long)blockIdx.x * 256 + threadIdx.x; if (tid >= total) return;
  float v = x[tid] + bias[tid % c];
  float t = 0.7978845608028654f * (v + 0.044715f * v * v * v);
  x[tid] = 0.5f * v * (1.f + tanhf(t));
}

__global__ void k_mean_last(const float* __restrict__ x, int C, int n,
                            float* __restrict__ out, long total)
{
  long tid = (long)blockIdx.x * 256 + threadIdx.x; if (tid >= total) return;
  const float* p = x + tid * (long)n;
  float s = 0.f;
  for (int j = 0; j < n; ++j) s += p[j];
  out[tid] = s / n;
}

__global__ void k_dense_relu(const float* __restrict__ W, const float* __restrict__ b,
                             const float* __restrict__ in, int Cin, int Cout,
                             float* __restrict__ out, long total)
{
  long tid = (long)blockIdx.x * 256 + threadIdx.x; if (tid >= total) return;
  int o = tid % Cout; int bb = tid / Cout;
  float s = b[o];
  for (int c = 0; c < Cin; ++c) s += W[(long)o * Cin + c] * in[(long)bb * Cin + c];
  out[tid] = fmaxf(s, 0.f);
}

__global__ void k_dense_sigmoid(const float* __restrict__ W, const float* __restrict__ b,
                                const float* __restrict__ in, int Cin, int Cout,
                                float* __restrict__ out, long total)
{
  long tid = (long)blockIdx.x * 256 + threadIdx.x; if (tid >= total) return;
  int o = tid % Cout; int bb = tid / Cout;
  float s = b[o];
  for (int c = 0; c < Cin; ++c) s += W[(long)o * Cin + c] * in[(long)bb * Cin + c];
  out[tid] = 1.f / (1.f + expf(-s));
}

__global__ void k_scale_ch(float* __restrict__ x, const float* __restrict__ g,
                           int C, int n, long total)
{
  long tid = (long)blockIdx.x * 256 + threadIdx.x; if (tid >= total) return;
  long r = tid / n; int c = r % C; int b = r / C;
  x[tid] *= g[(long)b * C + c];
}

__global__ void k_knn3(const float* __restrict__ c1, const float* __restrict__ c2,
                       int n1, int n2, int* __restrict__ idx, float* __restrict__ w, long total)
{
  long tid = (long)blockIdx.x * 256 + threadIdx.x; if (tid >= total) return;
  int b = tid / n1, i = tid % n1;
  float px = c1[((long)b*n1 + i)*3], py = c1[((long)b*n1 + i)*3+1], pz = c1[((long)b*n1 + i)*3+2];
  float d0 = 1e30f, d1 = 1e30f, d2 = 1e30f; int i0 = 0, i1 = 0, i2 = 0;
  for (int j = 0; j < n2; ++j) {
    float dx = c2[((long)b*n2 + j)*3] - px;
    float dy = c2[((long)b*n2 + j)*3+1] - py;
    float dz = c2[((long)b*n2 + j)*3+2] - pz;
    float d = dx*dx + dy*dy + dz*dz;
    if (d < d0)      { d2=d1; i2=i1; d1=d0; i1=i0; d0=d; i0=j; }
    else if (d < d1) { d2=d1; i2=i1; d1=d; i1=j; }
    else if (d < d2) { d2=d; i2=j; }
  }
  float w0 = 1.f/(d0+1e-8f), w1 = 1.f/(d1+1e-8f), w2 = 1.f/(d2+1e-8f);
  float inv = 1.f/(w0+w1+w2);
  idx[3*tid] = i0; idx[3*tid+1] = i1; idx[3*tid+2] = i2;
  w[3*tid] = w0*inv; w[3*tid+1] = w1*inv; w[3*tid+2] = w2*inv;
}

__global__ void k_fp_build(const float* __restrict__ pts1, int C1,
                           const float* __restrict__ pts2, int C2,
                           const int* __restrict__ idx, const float* __restrict__ w,
                           int n1, int n2, float* __restrict__ out, long total)
{
  long tid = (long)blockIdx.x * 256 + threadIdx.x; if (tid >= total) return;
  int C = C1 + C2;
  int i = tid % n1; long r = tid / n1; int c = r % C; int b = r / C;
  float v;
  if (c < C1) v = pts1[((long)b * C1 + c) * n1 + i];
  else {
    int cc = c - C1;
    long kb = ((long)b * n1 + i) * 3;
    const float* p = pts2 + ((long)b * C2 + cc) * n2;
    v = w[kb] * p[idx[kb]] + w[kb+1] * p[idx[kb+1]] + w[kb+2] * p[idx[kb+2]];
  }
  out[tid] = v;
}

__global__ void k_head_final(const float* __restrict__ z, const float* __restrict__ bias,
                             float* __restrict__ out, long total)
{
  long tid = (long)blockIdx.x * 256 + threadIdx.x; if (tid >= total) return;
  int b = tid / NPTS, nn = tid % NPTS;
  float p[NCLS];
  float m = -1e30f;
  for (int o = 0; o < NCLS; ++o) {
    p[o] = z[((long)b * NCLS + o) * NPTS + nn] + bias[o];
    m = fmaxf(m, p[o]);
  }
  float s = 0.f;
  for (int o = 0; o < NCLS; ++o) s += expf(p[o] - m);
  float ls = logf(s);
  float* pred = out;
  float* po   = out + (long)BATCH * NPTS * NCLS;
  long base = ((long)b * NPTS + nn) * NCLS;
  for (int o = 0; o < NCLS; ++o) {
    po[base + o]   = p[o];
    pred[base + o] = p[o] - m - ls;
  }
}

// ---------------------------------------------------------------------------
// host-side orchestration helpers
// ---------------------------------------------------------------------------
struct MlpLayer { const float* W; const float* g; const float* be; int cout; };

static void gemm(hipStream_t st, bool nt, const float* A, const float* B, float* C,
                 int M, int N, int K, int lda, int ldb, int ldc,
                 long sA, long sB, long sC, int batches)
{
  dim3 g((N + 63) / 64, (M + 15) / 16, batches);
  if (nt) wmma_gemm<true ><<<g, 32, 0, st>>>(A, B, C, M, N, K, lda, ldb, ldc, sA, sB, sC);
  else    wmma_gemm<false><<<g, 32, 0, st>>>(A, B, C, M, N, K, lda, ldb, ldc, sA, sB, sC);
}

static void conv_bn_relu(hipStream_t st, const float* W, const float* g, const float* be,
                         int cout, int cin, long Msp, const float* in,
                         float* zbuf, float* outp, float* stats)
{
  gemm(st, false, W, in, zbuf, cout, (int)Msp, cin,
       cin, (int)Msp, (int)Msp, 0, (long)cin * Msp, (long)cout * Msp, BATCH);
  bn_stats<<<cout, 256, 0, st>>>(zbuf, cout, Msp, (long)cout * Msp, stats);
  long total = (long)BATCH * cout * Msp;
  k_bn_apply_relu<<<(total + 255) / 256, 256, 0, st>>>(zbuf, stats, g, be, cout, Msp, outp, total);
}

static void sa_stage(hipStream_t st, const float* coords, int n, const float* feats, int Cf,
                     int np, float r2, int kk, const MlpLayer* L, int nl,
                     float* coordsOut, float* featsOut,
                     int* fpsIdx, int* ballIdx, float* sA, float* sB, float* stats)
{
  fps_kernel<<<BATCH, 256, 0, st>>>(coords, n, np, fpsIdx);
  long tg = (long)BATCH * np * 3;
  k_gather_coords<<<(tg + 255) / 256, 256, 0, st>>>(coords, fpsIdx, n, np, coordsOut, tg);
  long tb = (long)BATCH * np;
  k_ball<<<(tb + 255) / 256, 256, 0, st>>>(coords, coordsOut, n, np, kk, r2, ballIdx, tb);
  int C0 = 3 + Cf;
  long tq = (long)BATCH * C0 * kk * np;
  k_group<<<(tq + 255) / 256, 256, 0, st>>>(coords, coordsOut, feats, ballIdx, n, np, kk, Cf, sA, tq);
  const float* cur = sA; int cin = C0;
  for (int l = 0; l < nl; ++l) {
    float* z = (cur == sA) ? sB : sA;
    conv_bn_relu(st, L[l].W, L[l].g, L[l].be, L[l].cout, cin, (long)kk * np, cur, z, z, stats);
    cur = z; cin = L[l].cout;
  }
  long tm = (long)BATCH * cin * np;
  k_maxpool<<<(tm + 255) / 256, 256, 0, st>>>(cur, cin, kk, np, featsOut, tm);
}

static void fp_stage(hipStream_t st, const float* c1, int n1, const float* c2, int n2,
                     const float* pts1, int C1, const float* pts2, int C2,
                     const MlpLayer* L, int nl, float* outp,
                     int* knnIdx, float* knnW, float* sA, float* sB, float* stats)
{
  long tk = (long)BATCH * n1;
  k_knn3<<<(tk + 255) / 256, 256, 0, st>>>(c1, c2, n1, n2, knnIdx, knnW, tk);
  int C0 = C1 + C2;
  long tb = (long)BATCH * C0 * n1;
  k_fp_build<<<(tb + 255) / 256, 256, 0, st>>>(pts1, C1, pts2, C2, knnIdx, knnW, n1, n2, sA, tb);
  const float* cur = sA; int cin = C0;
  for (int l = 0; l < nl; ++l) {
    float* z = (cur == sA) ? sB : sA;
    float* o = (l == nl - 1) ? outp : z;
    conv_bn_relu(st, L[l].W, L[l].g, L[l].be, L[l].cout, cin, (long)n1, cur, z, o, stats);
    cur = o; cin = L[l].cout;
  }
}

static void tf_block(hipStream_t st, float* x, int s, int dim,
                     const float* ln1g, const float* ln1b, const float* Wqkv,
                     const float* Wo, const float* bo,
                     const float* ln2g, const float* ln2b,
                     const float* W1, const float* b1, const float* W2, const float* b2,
                     float* scratch)
{
  float* xsd  = scratch;
  float* lnb  = scratch +   600000;
  float* qkv  = scratch +  1200000;
  float* q    = scratch +  4500000;
  float* kbuf = scratch +  5600000;
  float* vbuf = scratch +  6700000;
  float* att  = scratch +  7800000;
  float* obuf = scratch + 12100000;
  float* osd  = scratch + 13200000;
  float* proj = scratch + 14300000;
  long tds = (long)BATCH * s * dim;
  k_cm2sd<<<(tds + 255) / 256, 256, 0, st>>>(x, dim, s, xsd, tds);
  k_ln<<<(BATCH * s + 255) / 256, 256, 0, st>>>(xsd, ln1g, ln1b, BATCH * s, dim, lnb);
  gemm(st, false, lnb, Wqkv, qkv, s, 1536, dim, dim, 1536, 1536,
       (long)s * dim, 0, (long)s * 1536, BATCH);
  long th = (long)BATCH * 8 * s * 64;
  k_split_heads<<<(th + 255) / 256, 256, 0, st>>>(qkv, s, q, kbuf, vbuf, th);
  gemm(st, true, q, kbuf, att, s, s, 64, 64, 64, s,
       (long)s * 64, (long)s * 64, (long)s * s, BATCH * 8);
  k_softmax<<<(BATCH * 8 * s + 255) / 256, 256, 0, st>>>(att, BATCH * 8 * s, s, 0.125f);
  gemm(st, false, att, vbuf, obuf, s, 64, s, s, 64, 64,
       (long)s * s, (long)s * 64, (long)s * 64, BATCH * 8);
  k_merge_heads<<<(th + 255) / 256, 256, 0, st>>>(obuf, s, osd, th);
  gemm(st, false, osd, Wo, proj, s, dim, 512, 512, dim, dim,
       (long)s * 512, 0, (long)s * dim, BATCH);
  k_add_bias_res<<<(tds + 255) / 256, 256, 0, st>>>(xsd, proj, bo, dim, tds);
  k_ln<<<(BATCH * s + 255) / 256, 256, 0, st>>>(xsd, ln2g, ln2b, BATCH * s, dim, lnb);
  gemm(st, false, lnb, W1, qkv, s, 64, dim, dim, 64, 64,
       (long)s * dim, 0, (long)s * 64, BATCH);
  long tgl = (long)BATCH * s * 64;
  k_gelu_bias<<<(tgl + 255) / 256, 256, 0, st>>>(qkv, b1, 64, tgl);
  gemm(st, false, qkv, W2, proj, s, dim, 64, 64, dim, dim,
       (long)s * 64, 0, (long)s * dim, BATCH);
  k_add_bias_res<<<(tds + 255) / 256, 256, 0, st>>>(xsd, proj, b2, dim, tds);
  k_sd2cm<<<(tds + 255) / 256, 256, 0, st>>>(xsd, dim, s, x, tds);
}

static void gbcf_block(hipStream_t st, float* x, int C, int n,
                       const float* W1, const float* b1, const float* W2, const float* b2,
                       float* cvec, float* g1, float* gv)
{
  int C4 = C / 4;
  k_mean_last<<<(BATCH * C + 255) / 256, 256, 0, st>>>(x, C, n, cvec, BATCH * C);
  k_dense_relu<<<(BATCH * C4 + 255) / 256, 256, 0, st>>>(W1, b1, cvec, C, C4, g1, BATCH * C4);
  k_dense_sigmoid<<<(BATCH * C + 255) / 256, 256, 0, st>>>(W2, b2, g1, C4, C, gv, BATCH * C);
  long ts = (long)BATCH * C * n;
  k_scale_ch<<<(ts + 255) / 256, 256, 0, st>>>(x, gv, C, n, ts);
}

// ---------------------------------------------------------------------------
extern "C" void kernel_launch(void* const* d_in, const int* in_sizes, int n_in,
                              void* d_out, int out_size, void* d_ws, size_t ws_size,
                              hipStream_t stream)
{
  (void)in_sizes; (void)n_in; (void)out_size; (void)ws_size;
#define IN(i) ((const float*)d_in[(i)])

  char* ws = (char*)d_ws;
  size_t off = 0;
  auto allocf = [&](size_t n) -> float* {
    float* p = (float*)(ws + off); off += n * 4; off = (off + 255) & ~(size_t)255; return p;
  };
  auto alloci = [&](size_t n) -> int* {
    int* p = (int*)(ws + off); off += n * 4; off = (off + 255) & ~(size_t)255; return p;
  };

  float* xyz_pm = allocf((size_t)BATCH * NPTS * 3);
  float* l1x = allocf((size_t)BATCH * 1024 * 3);
  float* l2x = allocf((size_t)BATCH * 256 * 3);
  float* l3x = allocf((size_t)BATCH * 64 * 3);
  float* l4x = allocf((size_t)BATCH * 16 * 3);
  float* l1p = allocf((size_t)BATCH * 128 * 1024);
  float* l2p = allocf((size_t)BATCH * 256 * 256);
  float* l3p = allocf((size_t)BATCH * 256 * 64);
  float* l4p = allocf((size_t)BATCH * 512 * 16);
  float* ms   = allocf((size_t)BATCH * 128 * NPTS);
  float* hbuf = allocf((size_t)BATCH * 128 * NPTS);
  float* stats = allocf(2048);
  float* cvec = allocf(BATCH * 512);
  float* g1b  = allocf(BATCH * 128);
  float* gvb  = allocf(BATCH * 512);
  int* fpsIdx  = alloci((size_t)BATCH * 1024);
  int* ballIdx = alloci((size_t)BATCH * 1024 * 32);
  int* knnIdx  = alloci((size_t)BATCH * NPTS * 3);
  float* knnW  = allocf((size_t)BATCH * NPTS * 3);
  float* sA = allocf(16777216);
  float* sB = allocf(16777216);

  // l0 coords (point-major)
  long t0 = (long)BATCH * NPTS * 3;
  k_strip_xyz<<<(t0 + 255) / 256, 256, 0, stream>>>(IN(0), xyz_pm, t0);

  // set abstraction stages
  MlpLayer sa1L[3] = {{IN(1), IN(3), IN(4), 32}, {IN(5), IN(7), IN(8), 32}, {IN(9), IN(11), IN(12), 64}};
  sa_stage(stream, xyz_pm, NPTS, IN(0), 9, 1024, 100.f, 32, sa1L, 3, l1x, l1p, fpsIdx, ballIdx, sA, sB, stats);
  MlpLayer sa2L[3] = {{IN(13), IN(15), IN(16), 64}, {IN(17), IN(19), IN(20), 64}, {IN(21), IN(23), IN(24), 128}};
  sa_stage(stream, l1x, 1024, l1p, 64, 256, 1600.f, 32, sa2L, 3, l2x, l2p, fpsIdx, ballIdx, sA, sB, stats);
  MlpLayer sa3L[3] = {{IN(25), IN(27), IN(28), 128}, {IN(29), IN(31), IN(32), 128}, {IN(33), IN(35), IN(36), 256}};
  sa_stage(stream, l2x, 256, l2p, 128, 64, 6400.f, 32, sa3L, 3, l3x, l3p, fpsIdx, ballIdx, sA, sB, stats);
  MlpLayer sa4L[3] = {{IN(37), IN(39), IN(40), 256}, {IN(41), IN(43), IN(44), 256}, {IN(45), IN(47), IN(48), 512}};
  sa_stage(stream, l3x, 64, l3p, 256, 16, 10000.f, 32, sa4L, 3, l4x, l4p, fpsIdx, ballIdx, sA, sB, stats);

  // tf4 + gbcf4 on l4p (dim 512, s 16)
  tf_block(stream, l4p, 16, 512, IN(49), IN(50), IN(51), IN(52), IN(53),
           IN(54), IN(55), IN(56), IN(57), IN(58), IN(59), sA);
  gbcf_block(stream, l4p, 512, 16, IN(60), IN(61), IN(62), IN(63), cvec, g1b, gvb);

  // fp4 -> l3p (768 -> 256 -> 256)
  MlpLayer fp4L[2] = {{IN(64), IN(66), IN(67), 256}, {IN(68), IN(70), IN(71), 256}};
  fp_stage(stream, l3x, 64, l4x, 16, l3p, 256, l4p, 512, fp4L, 2, l3p, knnIdx, knnW, sA, sB, stats);
  tf_block(stream, l3p, 64, 256, IN(72), IN(73), IN(74), IN(75), IN(76),
           IN(77), IN(78), IN(79), IN(80), IN(81), IN(82), sA);
  gbcf_block(stream, l3p, 256, 64, IN(83), IN(84), IN(85), IN(86), cvec, g1b, gvb);

  // fp3 -> l2p (384 -> 256 -> 256)
  MlpLayer fp3L[2] = {{IN(87), IN(89), IN(90), 256}, {IN(91), IN(93), IN(94), 256}};
  fp_stage(stream, l2x, 256, l3x, 64, l2p, 128, l3p, 256, fp3L, 2, l2p, knnIdx, knnW, sA, sB, stats);
  tf_block(stream, l2p, 256, 256, IN(95), IN(96), IN(97), IN(98), IN(99),
           IN(100), IN(101), IN(102), IN(103), IN(104), IN(105), sA);
  gbcf_block(stream, l2p, 256, 256, IN(106), IN(107), IN(108), IN(109), cvec, g1b, gvb);

  // fp2 -> l1p (320 -> 256 -> 128)
  MlpLayer fp2L[2] = {{IN(110), IN(112), IN(113), 256}, {IN(114), IN(116), IN(117), 128}};
  fp_stage(stream, l1x, 1024, l2x, 256, l1p, 64, l2p, 256, fp2L, 2, l1p, knnIdx, knnW, sA, sB, stats);

  // fp1 -> ms (128 -> 128 -> 128 -> 128), no pts1
  MlpLayer fp1L[3] = {{IN(118), IN(120), IN(121), 128}, {IN(122), IN(124), IN(125), 128}, {IN(126), IN(128), IN(129), 128}};
  fp_stage(stream, xyz_pm, NPTS, l1x, 1024, (const float*)nullptr, 0, l1p, 128, fp1L, 3, ms, knnIdx, knnW, sA, sB, stats);

  // head1: 128 -> 128 conv + BN + relu
  conv_bn_relu(stream, IN(130), IN(132), IN(133), 128, 128, (long)NPTS, ms, sA, hbuf, stats);
  // head2: 13 x 128 GEMM, then bias + log-softmax into d_out (pred, pred_o)
  gemm(stream, false, IN(134), hbuf, sB, NCLS, NPTS, 128, 128, NPTS, NPTS,
       0, (long)128 * NPTS, (long)NCLS * NPTS, BATCH);
  long tf_total = (long)BATCH * NPTS;
  k_head_final<<<(tf_total + 255) / 256, 256, 0, stream>>>(sB, IN(135), (float*)d_out, tf_total);
#undef IN
}